// Attn_34033320854092
// MI455X (gfx1250) — compile-verified
//
#include <hip/hip_runtime.h>
#include <hip/hip_bf16.h>
#include <math.h>

// Problem constants (from the reference)
#define TT 2048
#define BB 64
#define HH 1024

typedef __attribute__((ext_vector_type(2))) float v2f;
typedef __attribute__((ext_vector_type(4))) float v4f;
typedef __attribute__((ext_vector_type(8))) float v8f;

// ---------------------------------------------------------------------------
// Phase 1:  v[b,h] = sum_g hidden[b,g] * W[g,h]      (64x1024 @ 1024x1024)
// One wave per 16x16 output tile, K-loop in steps of 4 using f32 WMMA.
// 4 b-tiles x 64 h-tiles = 256 waves = 32 blocks of 256 threads.
// ---------------------------------------------------------------------------
__global__ void __launch_bounds__(256) attn_gemm_hW(const float* __restrict__ hidden,
                                                    const float* __restrict__ W,
                                                    float* __restrict__ v_out)
{
    const int lane = threadIdx.x & 31;
    const int wave = blockIdx.x * (blockDim.x >> 5) + (threadIdx.x >> 5);

    const int b_tile = wave >> 6;     // [0,4)
    const int h_tile = wave & 63;     // [0,64)
    const int b0 = b_tile * 16;
    const int h0 = h_tile * 16;

    const int half = lane >> 4;       // 0: lanes 0-15, 1: lanes 16-31
    const int m    = lane & 15;

    // A (16x4, f32): lanes 0-15: v0=K0, v1=K1 ; lanes 16-31: v0=K2, v1=K3
    const float* arow = hidden + (size_t)(b0 + m) * HH + 2 * half;
    // B (4x16, f32): lanes 0-15: v0=W[k+0][n], v1=W[k+1][n]; lanes 16-31: k+2/k+3
    const float* bcol = W + h0 + m;

    v8f acc = {};
    for (int k = 0; k < HH; k += 4) {
        v2f a = *(const v2f*)(arow + k);
        v2f bf;
        bf.x = bcol[(size_t)(k + 2 * half + 0) * HH];
        bf.y = bcol[(size_t)(k + 2 * half + 1) * HH];
        // D = A*B + C  (8 args: neg_a, A, neg_b, B, c_mod, C, reuse_a, reuse_b)
        acc = __builtin_amdgcn_wmma_f32_16x16x4_f32(false, a, false, bf,
                                                    (short)0, acc, false, false);
    }

    // D (16x16 f32): VGPR r holds row M = r + 8*half, column N = m
    float* drow = v_out + (size_t)(b0 + 8 * half) * HH + h0 + m;
#pragma unroll
    for (int r = 0; r < 8; ++r)
        drow[(size_t)r * HH] = acc[r];
}

// ---------------------------------------------------------------------------
// Phase 2:  scores[b,t] = enc[t,b,:] . v[b,:]   (streams 512 MiB once)
// One wave per (b, 16-t chunk).  v[b] is register-cached (8 x v4f / lane).
// Each enc row (4KB) is read with 8 coalesced non-temporal b128 loads per
// lane (read-once stream: TH_NT keeps it from evicting reused data).
// ---------------------------------------------------------------------------
__global__ void __launch_bounds__(256) attn_scores(const float* __restrict__ enc,
                                                   const float* __restrict__ v,
                                                   float* __restrict__ scores)
{
    const int TCHUNK = 16;
    const int lane = threadIdx.x & 31;
    const int wave = blockIdx.x * (blockDim.x >> 5) + (threadIdx.x >> 5);

    const int b     = wave >> 7;               // / (T/TCHUNK = 128)
    const int tbase = (wave & 127) * TCHUNK;

    // Cache v[b,:] in registers: lane holds elements {lane*4 + i*128 .. +3}
    const float* vb = v + (size_t)b * HH;
    v4f vr[8];
#pragma unroll
    for (int i = 0; i < 8; ++i)
        vr[i] = *(const v4f*)(vb + i * 128 + lane * 4);

    const size_t t_stride = (size_t)BB * HH;   // stride between t for fixed b

    for (int tt = 0; tt < TCHUNK; ++tt) {
        const int t = tbase + tt;
        const float* row = enc + ((size_t)t * BB + b) * HH;

        // Prefetch next row: 32 lanes x 128B lines cover the full 4KB row.
        if (tt + 1 < TCHUNK)
            __builtin_prefetch(row + t_stride + (size_t)lane * 32, 0, 1);

        v4f acc4 = {0.f, 0.f, 0.f, 0.f};
#pragma unroll
        for (int i = 0; i < 8; ++i) {
            v4f e = __builtin_nontemporal_load((const v4f*)(row + i * 128 + lane * 4));
            acc4.x = fmaf(e.x, vr[i].x, acc4.x);
            acc4.y = fmaf(e.y, vr[i].y, acc4.y);
            acc4.z = fmaf(e.z, vr[i].z, acc4.z);
            acc4.w = fmaf(e.w, vr[i].w, acc4.w);
        }
        float acc = (acc4.x + acc4.y) + (acc4.z + acc4.w);

        // wave32 butterfly reduction
#pragma unroll
        for (int off = 16; off > 0; off >>= 1)
            acc += __shfl_xor(acc, off, 32);

        if (lane == 0)
            scores[(size_t)b * TT + t] = acc;
    }
}

// ---------------------------------------------------------------------------
// Phase 3: in-place softmax over T=2048 per batch row.  One block per b.
// ---------------------------------------------------------------------------
__global__ void __launch_bounds__(256) attn_softmax(float* __restrict__ out)
{
    __shared__ float red[256];
    const int tid = threadIdx.x;
    float* row = out + (size_t)blockIdx.x * TT;

    float x[8];
    float m = -INFINITY;
#pragma unroll
    for (int j = 0; j < 8; ++j) {
        x[j] = row[tid + j * 256];
        m = fmaxf(m, x[j]);
    }
    red[tid] = m;
    __syncthreads();
    for (int s = 128; s > 0; s >>= 1) {
        if (tid < s) red[tid] = fmaxf(red[tid], red[tid + s]);
        __syncthreads();
    }
    const float M = red[0];
    __syncthreads();

    float sum = 0.f;
#pragma unroll
    for (int j = 0; j < 8; ++j) {
        x[j] = __expf(x[j] - M);
        sum += x[j];
    }
    red[tid] = sum;
    __syncthreads();
    for (int s = 128; s > 0; s >>= 1) {
        if (tid < s) red[tid] += red[tid + s];
        __syncthreads();
    }
    const float inv = 1.0f / red[0];
#pragma unroll
    for (int j = 0; j < 8; ++j)
        row[tid + j * 256] = x[j] * inv;
}

// ---------------------------------------------------------------------------
extern "C" void kernel_launch(void* const* d_in, const int* in_sizes, int n_in,
                              void* d_out, int out_size, void* d_ws, size_t ws_size,
                              hipStream_t stream)
{
    const float* hidden = (const float*)d_in[0]; // [1,B,H]
    const float* enc    = (const float*)d_in[1]; // [T,B,H]
    const float* W      = (const float*)d_in[2]; // [H,H]
    // d_in[3] = b_attn: contributes a per-b constant to scores -> cancels in softmax.

    float* out = (float*)d_out;   // [B,1,T] == [B,T] flat
    float* v   = (float*)d_ws;    // [B,H] = 256 KB scratch

    // Phase 1: 256 waves (32 blocks x 8 waves)
    hipLaunchKernelGGL(attn_gemm_hW, dim3(32), dim3(256), 0, stream, hidden, W, v);
    // Phase 2: 8192 waves (1024 blocks x 8 waves), streams enc once
    hipLaunchKernelGGL(attn_scores, dim3(1024), dim3(256), 0, stream, enc, v, out);
    // Phase 3: 64 blocks, softmax in place
    hipLaunchKernelGGL(attn_softmax, dim3(64), dim3(256), 0, stream, out);
}